// L0SINDy_reward_15350213116283
// MI455X (gfx1250) — compile-verified
//
#include <hip/hip_runtime.h>
#include <math.h>

// CDNA5 / gfx1250: wave32, WMMA (not MFMA).
typedef __attribute__((ext_vector_type(2))) float v2f;
typedef __attribute__((ext_vector_type(8))) float v8f;

#define HC_GAMMA (-0.1f)
#define HC_ZETA  (1.1f)

// Deterministic hard-concrete gate * weight, full f32.
__device__ __forceinline__ float hc_gate_coef(const float* __restrict__ qz,
                                              const float* __restrict__ w,
                                              int k) {
  float q = qz[k];
  float s = 1.0f / (1.0f + __expf(-q));
  float z = s * (HC_ZETA - HC_GAMMA) + HC_GAMMA;
  z = fminf(fmaxf(z, 0.0f), 1.0f);
  return z * w[k];
}

__global__ void __launch_bounds__(256)
l0sindy_poly_wmma(const float* __restrict__ obs,      // [B,3]
                  const float* __restrict__ act,      // [B,1]
                  const float* __restrict__ weights,  // [35,1]
                  const float* __restrict__ qz_loga,  // [35]
                  float* __restrict__ out,            // [B,1]
                  int batch) {
  const int lane          = threadIdx.x & 31;
  const int waveInBlock   = threadIdx.x >> 5;
  const int wavesPerBlock = blockDim.x >> 5;
  const int waveId        = blockIdx.x * wavesPerBlock + waveInBlock;
  const int numWaves      = gridDim.x * wavesPerBlock;
  const int numTiles      = batch >> 4;        // 16 rows per WMMA tile

  const bool hi    = lane >= 16;               // A-matrix K-half per ISA layout
  const int  row16 = lane & 15;                // M = lane % 16
  const int  ko    = hi ? 2 : 0;

  // ---- B matrix: gated coefficients, replicated across N columns. ----
  // Lane supplies B[k,*] for k = 4j+ko and 4j+ko+1 (k=35 is zero pad).
  float b0[9], b1[9];
#pragma unroll
  for (int j = 0; j < 9; ++j) {
    int k = 4 * j + ko;                        // max 34 -> always valid
    b0[j] = hc_gate_coef(qz_loga, weights, k);
    b1[j] = (k + 1 < 35) ? hc_gate_coef(qz_loga, weights, k + 1) : 0.0f;
  }

  for (int tile = waveId; tile < numTiles; tile += numWaves) {
    const int r  = (tile << 4) + row16;
    const float x0 = obs[3 * r + 0];
    const float x1 = obs[3 * r + 1];
    const float x2 = obs[3 * r + 2];
    const float x3 = act[r];

    // Prefetch next grid-stride tile (gfx1250 global_prefetch_b8).
    const int nt = tile + numWaves;
    if (nt < numTiles) {
      const int rn = (nt << 4) + row16;
      __builtin_prefetch(&obs[3 * rn], 0, 0);
      __builtin_prefetch(&act[rn], 0, 0);
    }

    // ---- Monomials, sklearn PolynomialFeatures(3) ordering (all scalars,
    // no local array -> everything stays in VGPRs, no LDS promotion). ----
    // degree 2 (10 products)
    const float p00 = x0 * x0, p01 = x0 * x1, p02 = x0 * x2, p03 = x0 * x3;
    const float p11 = x1 * x1, p12 = x1 * x2, p13 = x1 * x3;
    const float p22 = x2 * x2, p23 = x2 * x3, p33 = x3 * x3;
    // degree 3 (20 products, reusing degree 2)
    const float q000 = x0 * p00, q001 = x0 * p01, q002 = x0 * p02, q003 = x0 * p03;
    const float q011 = x0 * p11, q012 = x0 * p12, q013 = x0 * p13;
    const float q022 = x0 * p22, q023 = x0 * p23, q033 = x0 * p33;
    const float q111 = x1 * p11, q112 = x1 * p12, q113 = x1 * p13;
    const float q122 = x1 * p22, q123 = x1 * p23, q133 = x1 * p33;
    const float q222 = x2 * p22, q223 = x2 * p23, q233 = x2 * p33;
    const float q333 = x3 * p33;

    // ---- 9 accumulating f32 WMMAs: D[16x16] = sum_j A_j(16x4) x B_j(4x16).
    // Chunk j covers feature indices 4j..4j+3; low lanes feed K={0,1},
    // high lanes K={2,3} (ISA 16x4 f32 A layout) via scalar cndmask selects.
    v8f acc = {0.0f, 0.0f, 0.0f, 0.0f, 0.0f, 0.0f, 0.0f, 0.0f};
#define WMMA_STEP(t0, t1, t2, t3, j)                                       \
    {                                                                      \
      v2f A  = {hi ? (t2) : (t0), hi ? (t3) : (t1)};                       \
      v2f Bv = {b0[j], b1[j]};                                             \
      acc = __builtin_amdgcn_wmma_f32_16x16x4_f32(                         \
          false, A, false, Bv, (short)0, acc, false, false);               \
    }
    WMMA_STEP(1.0f, x0,   x1,   x2,   0)   // f0..f3
    WMMA_STEP(x3,   p00,  p01,  p02,  1)   // f4..f7
    WMMA_STEP(p03,  p11,  p12,  p13,  2)   // f8..f11
    WMMA_STEP(p22,  p23,  p33,  q000, 3)   // f12..f15
    WMMA_STEP(q001, q002, q003, q011, 4)   // f16..f19
    WMMA_STEP(q012, q013, q022, q023, 5)   // f20..f23
    WMMA_STEP(q033, q111, q112, q113, 6)   // f24..f27
    WMMA_STEP(q122, q123, q133, q222, 7)   // f28..f31
    WMMA_STEP(q223, q233, q333, 0.0f, 8)   // f32..f34 + zero pad
#undef WMMA_STEP

    // ---- Store: D column 0 is the result vector (replicated over N).
    // lane 0 holds rows 0..7, lane 16 holds rows 8..15 (ISA C/D layout).
    if (row16 == 0) {
      float* o = out + (tile << 4) + (hi ? 8 : 0);   // 32B aligned
      float4 v0 = make_float4(acc[0], acc[1], acc[2], acc[3]);
      float4 v1 = make_float4(acc[4], acc[5], acc[6], acc[7]);
      *reinterpret_cast<float4*>(o)     = v0;
      *reinterpret_cast<float4*>(o + 4) = v1;
    }
  }
}

extern "C" void kernel_launch(void* const* d_in, const int* in_sizes, int n_in,
                              void* d_out, int out_size, void* d_ws, size_t ws_size,
                              hipStream_t stream) {
  const float* obs     = (const float*)d_in[0];
  const float* act     = (const float*)d_in[1];
  const float* weights = (const float*)d_in[2];
  const float* qz_loga = (const float*)d_in[3];
  float* out = (float*)d_out;

  const int batch    = in_sizes[1];          // act element count == B
  const int numTiles = batch >> 4;
  const int threads  = 256;                  // 8 wave32s
  const int wavesPerBlock = threads / 32;
  int blocks = (numTiles + wavesPerBlock - 1) / wavesPerBlock;
  if (blocks > 4096) blocks = 4096;          // 32K waves, grid-stride over tiles
  if (blocks < 1) blocks = 1;

  l0sindy_poly_wmma<<<blocks, threads, 0, stream>>>(obs, act, weights, qz_loga,
                                                    out, batch);
}